// AttentionLayerPooler_59889023975482
// MI455X (gfx1250) — compile-verified
//
#include <hip/hip_runtime.h>

typedef __attribute__((ext_vector_type(2))) float v2f;
typedef __attribute__((ext_vector_type(8))) float v8f;

#define L_TEACHER 36
#define L_STUDENT 28
#define KSTEPS 9          // 36 / 4
#define A_ROWS 32         // 28 padded to 32 (two 16-row WMMA tiles)

// ---------------------------------------------------------------------------
// Kernel 1: softmax of both 28x36 logit matrices into d_ws, padded to 32 rows.
// ws layout: attn[2][32][36] floats (K first, then V).
// ---------------------------------------------------------------------------
__global__ void attn_softmax_kernel(const float* __restrict__ logits_k,
                                    const float* __restrict__ logits_v,
                                    float* __restrict__ attn) {
    const int m = threadIdx.x;            // 0..31, one wave
    for (int w = 0; w < 2; ++w) {
        const float* L = (w == 0) ? logits_k : logits_v;
        float* A = attn + w * A_ROWS * L_TEACHER;
        if (m < L_STUDENT) {
            const float* row = L + m * L_TEACHER;
            float mx = -3.402823e38f;
            for (int l = 0; l < L_TEACHER; ++l) mx = fmaxf(mx, row[l]);
            float s = 0.0f;
            for (int l = 0; l < L_TEACHER; ++l) s += __expf(row[l] - mx);
            const float inv = 1.0f / s;
            for (int l = 0; l < L_TEACHER; ++l)
                A[m * L_TEACHER + l] = __expf(row[l] - mx) * inv;
        } else if (m < A_ROWS) {
            for (int l = 0; l < L_TEACHER; ++l) A[m * L_TEACHER + l] = 0.0f;
        }
    }
}

// ---------------------------------------------------------------------------
// Kernel 2: Out[28, N] = A[28,36] x X[36, N] via V_WMMA_F32_16X16X4_F32.
// One wave computes a 32(M)x16(N) tile (rows 28..31 dropped at store time).
// 8 waves per block -> 128 contiguous columns per block.
// blockIdx.y: 0 = K path, 1 = V path.
// All per-lane offsets are 32-bit (tensor <= 288 MB) so addresses resolve to
// SADDR(64-bit uniform base) + 32-bit VGPR offset; row strides stay in SALU.
// ---------------------------------------------------------------------------
__global__ __launch_bounds__(256) void pool_wmma_kernel(
        const float* __restrict__ ks, const float* __restrict__ vs,
        const float* __restrict__ attn, float* __restrict__ out,
        unsigned int N) {
    const unsigned int tid    = threadIdx.x;
    const unsigned int wave   = tid >> 5;
    const unsigned int lane   = tid & 31u;
    const unsigned int lane16 = lane & 15u;
    const unsigned int hi     = lane >> 4;     // 0: lanes 0-15, 1: lanes 16-31
    const unsigned int which  = blockIdx.y;

    const float* __restrict__ X = (which == 0) ? ks : vs;
    const float* __restrict__ A = attn + which * A_ROWS * L_TEACHER;
    float* __restrict__ O = out + (size_t)which * (size_t)L_STUDENT * N;

    const unsigned int col = (blockIdx.x * 8u + wave) * 16u + lane16;

    // Preload A fragments. 16x4 fp32 A layout:
    //   lanes 0-15 : VGPR0=A[M][k0+0], VGPR1=A[M][k0+1]
    //   lanes 16-31: VGPR0=A[M][k0+2], VGPR1=A[M][k0+3]
    v2f a0[KSTEPS], a1[KSTEPS];
    {
        const unsigned int koff = hi * 2u;
        const float* Ar0 = A + lane16 * L_TEACHER;          // rows 0..15
        const float* Ar1 = A + (16u + lane16) * L_TEACHER;  // rows 16..31
#pragma unroll
        for (int k = 0; k < KSTEPS; ++k) {
            a0[k] = *(const v2f*)(Ar0 + 4 * k + koff);      // 8B aligned
            a1[k] = *(const v2f*)(Ar1 + 4 * k + koff);
        }
    }

    // Per-lane 32-bit load offset (computed once); uniform row strides added
    // to the scalar base pointer per K-step.
    const unsigned int laneOff = hi * 2u * N + col;

    v8f acc0 = {};
    v8f acc1 = {};
#pragma unroll
    for (int k = 0; k < KSTEPS; ++k) {
        // B fragment (4x16): VGPR0 lanes0-15=K0 / lanes16-31=K2; VGPR1 = K1/K3.
        const float* Xk0 = X + (size_t)(4 * k) * N;         // uniform (SALU)
        const float* Xk1 = Xk0 + N;                         // uniform (SALU)
        v2f b;
        b.x = __builtin_nontemporal_load(Xk0 + laneOff);
        b.y = __builtin_nontemporal_load(Xk1 + laneOff);
        acc0 = __builtin_amdgcn_wmma_f32_16x16x4_f32(
                   false, a0[k], false, b, (short)0, acc0, false, false);
        acc1 = __builtin_amdgcn_wmma_f32_16x16x4_f32(
                   false, a1[k], false, b, (short)0, acc1, false, false);
    }

    // C/D layout: VGPR r -> row r (lanes 0-15) / row r+8 (lanes 16-31).
    // Per-lane store offset once; uniform (row*N) folded into scalar base.
    const unsigned int sOff = hi * 8u * N + col;
#pragma unroll
    for (int r = 0; r < 8; ++r) {
        // tile 0: rows r / r+8 -> always < 28
        __builtin_nontemporal_store(acc0[r], O + (size_t)r * N + sOff);
        // tile 1: rows 16+r / 24+r -> 24+r valid only for r < 4
        if (r < 4) {
            __builtin_nontemporal_store(acc1[r], O + (size_t)(16 + r) * N + sOff);
        } else if (hi == 0) {
            __builtin_nontemporal_store(acc1[r], O + (size_t)(16 + r) * N + sOff);
        }
    }
}

// ---------------------------------------------------------------------------
extern "C" void kernel_launch(void* const* d_in, const int* in_sizes, int n_in,
                              void* d_out, int out_size, void* d_ws, size_t ws_size,
                              hipStream_t stream) {
    const float* ks = (const float*)d_in[0];
    const float* vs = (const float*)d_in[1];
    const float* lk = (const float*)d_in[2];
    const float* lv = (const float*)d_in[3];
    float* out  = (float*)d_out;
    float* attn = (float*)d_ws;                     // 2*32*36 floats = 9216 B

    const unsigned int Ncols = (unsigned int)(in_sizes[0] / L_TEACHER); // 2,097,152

    attn_softmax_kernel<<<1, 32, 0, stream>>>(lk, lv, attn);

    dim3 grid(Ncols / 128u, 2);
    pool_wmma_kernel<<<grid, 256, 0, stream>>>(ks, vs, attn, out, Ncols);
}